// Yolo_45045617001025
// MI455X (gfx1250) — compile-verified
//
#include <hip/hip_runtime.h>
#include <stdint.h>

// ---------------- problem constants ----------------
#define BS       16
#define NANCH    25200
#define NCLS     80
#define CSTRIDE  85          // xy(2) wh(2) obj(1) cls(80)
#define KNMS     2048
#define CAP      4096
#define NBUCKET  2048
#define MAXDET   300
#define CONF_T   0.25f
#define IOU_T    0.45f
#define MAX_WH   7680.0f
#define SCALE_X  3.0f        // 1920/640
#define SCALE_Y  1.6875f     // 1080/640

typedef __attribute__((ext_vector_type(2))) float v2f;
typedef __attribute__((ext_vector_type(8))) float v8f;

// ---------------- CDNA5 async LDS DMA helpers ----------------
typedef __attribute__((address_space(3))) void lds_void;

__device__ __forceinline__ void async_copy16(void* lds_dst, const void* gsrc) {
  // VDST = 32-bit LDS byte address, VADDR = 64-bit global address (GV mode).
  unsigned l = (unsigned)(unsigned long long)(lds_void*)lds_dst;
  asm volatile("global_load_async_to_lds_b128 %0, %1, off"
               :: "v"(l), "v"(gsrc) : "memory");
}
__device__ __forceinline__ void wait_async0() {
  asm volatile("s_wait_asynccnt 0x0" ::: "memory");
}

__device__ __forceinline__ float iou4(float4 a, float4 b) {
  float lx = fmaxf(a.x, b.x), ly = fmaxf(a.y, b.y);
  float rx = fminf(a.z, b.z), ry = fminf(a.w, b.w);
  float w = fmaxf(rx - lx, 0.0f), h = fmaxf(ry - ly, 0.0f);
  float inter = w * h;
  float a1 = (a.z - a.x) * (a.w - a.y);
  float a2 = (b.z - b.x) * (b.w - b.y);
  return inter / (a1 + a2 - inter + 1e-7f);
}

// ---------------- kernel 0: zero scratch + output ----------------
__global__ void zeroKernel(int* hist, int* cnt, float* out, int outN) {
  int i = blockIdx.x * 256 + threadIdx.x;
  if (i < BS * NBUCKET) hist[i] = 0;
  if (i < BS)           cnt[i]  = 0;
  if (i < outN)         out[i]  = 0.0f;
}

// ---------------- kernel 1: score/class + histogram (bandwidth kernel) ----
#define APB 128   // anchors per block (must be multiple of 4 for 16B chunking)
__global__ __launch_bounds__(APB)
void scoreKernel(const float* __restrict__ preds,
                 float* __restrict__ score, int* __restrict__ clsOut,
                 int* __restrict__ hist) {
  const int b  = blockIdx.y;
  const int a0 = blockIdx.x * APB;
  const int nA = min(APB, NANCH - a0);              // 128 or 112, both %4==0
  __shared__ __align__(16) float tile[APB * CSTRIDE];

  const char* g = (const char*)(preds + ((size_t)b * NANCH + a0) * CSTRIDE);
  const int nCh = nA * CSTRIDE / 4;                 // 16-byte chunks
  for (int c = threadIdx.x; c < nCh; c += APB)
    async_copy16(&tile[c * 4], g + (size_t)c * 16);
  wait_async0();
  __syncthreads();

  if ((int)threadIdx.x < nA) {
    const float* p = &tile[threadIdx.x * CSTRIDE];
    float obj = p[4];
    float best = -1.0f; int bc = 0;
    #pragma unroll 8
    for (int c = 0; c < NCLS; ++c) {
      float v = p[5 + c] * obj;
      if (v > best) { best = v; bc = c; }            // first-max == argmax
    }
    bool m = (obj > CONF_T) && (best > CONF_T);
    float s = m ? best : -1.0f;
    size_t o = (size_t)b * NANCH + (a0 + threadIdx.x);
    score[o]  = s;
    clsOut[o] = bc;
    if (s > 0.0f) {
      int bk = (int)(s * (float)NBUCKET); if (bk > NBUCKET - 1) bk = NBUCKET - 1;
      atomicAdd(&hist[b * NBUCKET + bk], 1);
    }
  }
}

// ---------------- kernel 2: per-batch score threshold bucket --------------
__global__ __launch_bounds__(256)
void threshKernel(const int* __restrict__ hist, int* __restrict__ iThresh,
                  int* __restrict__ cnt) {
  const int b = blockIdx.x;
  __shared__ int h[NBUCKET];
  for (int t = threadIdx.x; t < NBUCKET; t += 256) h[t] = hist[b * NBUCKET + t];
  __syncthreads();
  if (threadIdx.x == 0) {
    int acc = 0, T = 0;
    for (int t = NBUCKET - 1; t >= 0; --t) {
      acc += h[t];
      if (acc >= KNMS) { T = t; break; }
    }
    iThresh[b] = T;     // T==0 => take every positive (fewer than K exist)
    cnt[b] = 0;
  }
}

// ---------------- kernel 3: compact candidates ----------------------------
__global__ __launch_bounds__(256)
void compactKernel(const float* __restrict__ score, const int* __restrict__ iThresh,
                   int* __restrict__ cnt, int* __restrict__ candIdx,
                   float* __restrict__ candScoreC) {
  const int b = blockIdx.y;
  const int n = blockIdx.x * 256 + threadIdx.x;
  if (n >= NANCH) return;
  float s = score[(size_t)b * NANCH + n];
  if (s > 0.0f) {
    int bk = (int)(s * (float)NBUCKET); if (bk > NBUCKET - 1) bk = NBUCKET - 1;
    if (bk >= iThresh[b]) {
      int pos = atomicAdd(&cnt[b], 1);
      if (pos < CAP) {
        candIdx[b * CAP + pos]    = n;
        candScoreC[b * CAP + pos] = s;
      }
    }
  }
}

// ---------------- kernel 4: bitonic top-K sort + WMMA box transform -------
// D(16x16) = A(16x4:[x,y,w,h] of 16 boxes) x B(4x16) + 0
//   B cols 0..3: xywh->xyxy (NMS box; class offset added at store time)
//   B cols 4..7: xywh->xyxy scaled to output resolution
__global__ __launch_bounds__(1024)
void sortKernel(const int* __restrict__ cnt, const int* __restrict__ candIdx,
                const float* __restrict__ candScoreC,
                const float* __restrict__ preds, const int* __restrict__ clsArr,
                float* __restrict__ candScaled, float* __restrict__ candBoxOff,
                float* __restrict__ candScoreK, int* __restrict__ candClsK) {
  const int b = blockIdx.x;
  __shared__ unsigned long long key[CAP];   // 32KB
  __shared__ int   sIdx[KNMS];              // 8KB  (-1 = invalid/pad)
  __shared__ float sClsF[KNMS];             // 8KB
  int n = cnt[b]; if (n > CAP) n = CAP;

  // key = score_bits(31-bit positive float) << 32 | ~idx  -> sort descending
  for (int t = threadIdx.x; t < CAP; t += 1024) {
    unsigned long long k = 0ull;
    if (t < n) {
      unsigned sb = __float_as_uint(candScoreC[b * CAP + t]);
      unsigned id = (unsigned)candIdx[b * CAP + t];
      k = ((unsigned long long)sb << 32) | (unsigned)(~id);
    }
    key[t] = k;
  }
  __syncthreads();

  for (int kk = 2; kk <= CAP; kk <<= 1) {
    for (int j = kk >> 1; j > 0; j >>= 1) {
      for (int t = threadIdx.x; t < CAP; t += 1024) {
        int ixj = t ^ j;
        if (ixj > t) {
          bool desc = ((t & kk) == 0);
          unsigned long long a = key[t], c = key[ixj];
          if (desc ? (a < c) : (a > c)) { key[t] = c; key[ixj] = a; }
        }
      }
      __syncthreads();
    }
  }

  // Phase 1: decode keys, emit score/class, stage idx+cls in LDS.
  for (int t = threadIdx.x; t < KNMS; t += 1024) {
    unsigned long long k = key[t];
    int   idx = -1, c = 0; float s = 0.0f;
    if (k != 0ull) {
      s   = __uint_as_float((unsigned)(k >> 32));
      idx = (int)(~(unsigned)(k & 0xFFFFFFFFull));
      c   = clsArr[(size_t)b * NANCH + idx];
    }
    sIdx[t]  = idx;
    sClsF[t] = (float)c;
    candScoreK[(size_t)b * KNMS + t] = s;
    candClsK[(size_t)b * KNMS + t]   = c;
  }
  __syncthreads();

  // Phase 2: WMMA transform, 16 boxes per wave, 4 passes (32 waves x 16 x 4 = 2048).
  const int wave = threadIdx.x >> 5;
  const int lane = threadIdx.x & 31;
  const bool hi  = lane >= 16;      // lanes 16..31 carry K=2,3 (A) / rows 8..15 (D)
  const int nn   = lane & 15;       // column index for B and D
  const float* pb = preds + (size_t)b * NANCH * CSTRIDE;

  // B matrix, rows K0..K3 at column nn (see header comment).
  float t0 = 0.f, t1 = 0.f, t2 = 0.f, t3 = 0.f;
  if (nn < 8) {
    bool odd = (nn & 1);
    bool far = (nn >= 4);
    bool plus = (nn & 2);
    if (!odd) { t0 = far ? SCALE_X : 1.0f;                    // x contribution
                t2 = (plus ? 0.5f : -0.5f) * (far ? SCALE_X : 1.0f); }
    else      { t1 = far ? SCALE_Y : 1.0f;                    // y contribution
                t3 = (plus ? 0.5f : -0.5f) * (far ? SCALE_Y : 1.0f); }
  }
  v2f bm; bm.x = hi ? t2 : t0; bm.y = hi ? t3 : t1;

  for (int base = wave * 16; base < KNMS; base += 32 * 16) {
    int m   = base + nn;            // this lane's A-row (box)
    int idx = sIdx[m];
    v2f am; am.x = 0.f; am.y = 0.f;
    if (idx >= 0) {
      const float* p = pb + (size_t)idx * CSTRIDE;
      am.x = hi ? p[2] : p[0];      // K0:x / K2:w
      am.y = hi ? p[3] : p[1];      // K1:y / K3:h
    }
    v8f cm = {0.f, 0.f, 0.f, 0.f, 0.f, 0.f, 0.f, 0.f};
    v8f d = __builtin_amdgcn_wmma_f32_16x16x4_f32(
        /*neg_a=*/false, am, /*neg_b=*/false, bm,
        /*c_mod=*/(short)0, cm, /*reuse_a=*/false, /*reuse_b=*/false);

    if (nn < 8) {
      const int mOff = hi ? 8 : 0;  // D rows: r + (hi?8:0)
      #pragma unroll
      for (int r = 0; r < 8; ++r) {
        int box = base + r + mOff;
        size_t o4 = ((size_t)b * KNMS + box) * 4;
        if (nn < 4) candBoxOff[o4 + nn]        = d[r] + MAX_WH * sClsF[box];
        else        candScaled[o4 + (nn - 4)]  = d[r];
      }
    }
  }
}

// ---------------- kernel 5: blocked greedy NMS + emit top-300 -------------
__global__ __launch_bounds__(1024)
void nmsKernel(const float* __restrict__ candBoxOff, const float* __restrict__ candScoreK,
               const float* __restrict__ candScaled, const int* __restrict__ candClsK,
               float* __restrict__ out) {
  const int b = blockIdx.x;
  __shared__ __align__(16) float sBox[KNMS * 4];     // class-offset xyxy
  __shared__ __align__(16) float sScore[KNMS];
  __shared__ unsigned sSupp[KNMS];                   // suppressed flags, later scan buf
  __shared__ unsigned sKeep[KNMS];
  __shared__ unsigned sMask[130];                    // 64 rows x 2 words + kept mask

  // Async-DMA boxes (32KB) + scores (8KB) into LDS.
  const char* gb = (const char*)(candBoxOff + (size_t)b * KNMS * 4);
  const char* gs = (const char*)(candScoreK + (size_t)b * KNMS);
  for (int c = threadIdx.x; c < KNMS; c += 1024)        // 2048 x 16B
    async_copy16(&sBox[c * 4], gb + (size_t)c * 16);
  for (int c = threadIdx.x; c < KNMS / 4; c += 1024)    // 512 x 16B
    async_copy16(&sScore[c * 4], gs + (size_t)c * 16);
  for (int t = threadIdx.x; t < KNMS; t += 1024) { sSupp[t] = 0u; sKeep[t] = 0u; }
  wait_async0();
  __syncthreads();

  const float4* sB4 = (const float4*)sBox;

  for (int blk = 0; blk < KNMS / 64; ++blk) {
    const int base = blk * 64;
    if (threadIdx.x < 130) sMask[threadIdx.x] = 0u;
    __syncthreads();

    // 1) 64x64 intra-block IoU bitmask (forward pairs only: c > r)
    for (int p = threadIdx.x; p < 4096; p += 1024) {
      int r = p >> 6, c = p & 63;
      if (c > r) {
        if (iou4(sB4[base + r], sB4[base + c]) > IOU_T)
          atomicOr(&sMask[r * 2 + (c >> 5)], 1u << (c & 31));
      }
    }
    __syncthreads();

    // 2) sequential resolution of this block (the only serial dependency)
    if (threadIdx.x == 0) {
      unsigned lo = 0u, hi = 0u, kl = 0u, kh = 0u;
      for (int k = 0; k < 64; ++k) {
        int i = base + k;
        bool lsup = (k < 32) ? ((lo >> k) & 1u) : ((hi >> (k - 32)) & 1u);
        bool keepi = (sScore[i] > 0.0f) && (sSupp[i] == 0u) && !lsup;
        sKeep[i] = keepi ? 1u : 0u;
        if (keepi) {
          lo |= sMask[k * 2]; hi |= sMask[k * 2 + 1];
          if (k < 32) kl |= 1u << k; else kh |= 1u << (k - 32);
        }
      }
      sMask[128] = kl; sMask[129] = kh;
    }
    __syncthreads();

    // 3) kept boxes of this block suppress all later candidates in parallel
    unsigned kl = sMask[128], kh = sMask[129];
    if (kl | kh) {
      for (int j = base + 64 + (int)threadIdx.x; j < KNMS; j += 1024) {
        if (sSupp[j] == 0u) {
          bool sup = false;
          unsigned m = kl;
          while (m && !sup) { int k = __ffs(m) - 1; m &= m - 1;
            if (iou4(sB4[base + k], sB4[j]) > IOU_T) sup = true; }
          m = kh;
          while (m && !sup) { int k = __ffs(m) + 31; m &= m - 1;
            if (iou4(sB4[base + k], sB4[j]) > IOU_T) sup = true; }
          if (sup) sSupp[j] = 1u;
        }
      }
    }
    __syncthreads();
  }

  // Inclusive scan of keep flags (Hillis-Steele, sSupp reused as scan buffer)
  for (int t = threadIdx.x; t < KNMS; t += 1024) sSupp[t] = sKeep[t];
  __syncthreads();
  for (int off = 1; off < KNMS; off <<= 1) {
    int t0 = threadIdx.x, t1 = threadIdx.x + 1024;
    unsigned v0 = sSupp[t0] + ((t0 >= off) ? sSupp[t0 - off] : 0u);
    unsigned v1 = sSupp[t1] + ((t1 >= off) ? sSupp[t1 - off] : 0u);
    __syncthreads();
    sSupp[t0] = v0; sSupp[t1] = v1;
    __syncthreads();
  }

  // Emit top-300 kept detections (boxes pre-scaled by the WMMA transform).
  for (int t = threadIdx.x; t < KNMS; t += 1024) {
    if (sKeep[t]) {
      int rank = (int)sSupp[t] - 1;        // exclusive rank
      if (rank < MAXDET) {
        const float* bx = candScaled + ((size_t)b * KNMS + t) * 4;
        float s = sScore[t];
        int   c = candClsK[(size_t)b * KNMS + t];
        float* o = out + ((size_t)b * MAXDET + rank) * 6;
        o[0] = bx[0]; o[1] = bx[1]; o[2] = bx[2]; o[3] = bx[3];
        o[4] = s;     o[5] = (float)c;
      }
    }
  }
}

// ---------------- host side ----------------
extern "C" void kernel_launch(void* const* d_in, const int* in_sizes, int n_in,
                              void* d_out, int out_size, void* d_ws, size_t ws_size,
                              hipStream_t stream) {
  const float* preds = (const float*)d_in[0];
  float* out = (float*)d_out;

  char* ws = (char*)d_ws;
  size_t o = 0;
  auto alloc = [&](size_t bytes) -> void* {
    void* p = ws + o; o += (bytes + 255) & ~(size_t)255; return p;
  };
  float* score      = (float*)alloc((size_t)BS * NANCH * 4);
  int*   clsArr     = (int*)  alloc((size_t)BS * NANCH * 4);
  int*   hist       = (int*)  alloc((size_t)BS * NBUCKET * 4);
  int*   iThresh    = (int*)  alloc((size_t)BS * 4);
  int*   cnt        = (int*)  alloc((size_t)BS * 4);
  int*   candIdx    = (int*)  alloc((size_t)BS * CAP * 4);
  float* candScoreC = (float*)alloc((size_t)BS * CAP * 4);
  float* candScaled = (float*)alloc((size_t)BS * KNMS * 4 * 4);
  float* candBoxOff = (float*)alloc((size_t)BS * KNMS * 4 * 4);
  float* candScoreK = (float*)alloc((size_t)BS * KNMS * 4);
  int*   candClsK   = (int*)  alloc((size_t)BS * KNMS * 4);
  (void)ws_size; (void)in_sizes; (void)n_in;

  zeroKernel<<<(BS * NBUCKET + 255) / 256, 256, 0, stream>>>(hist, cnt, out, out_size);

  dim3 gA((NANCH + APB - 1) / APB, BS);
  scoreKernel<<<gA, APB, 0, stream>>>(preds, score, clsArr, hist);

  threshKernel<<<BS, 256, 0, stream>>>(hist, iThresh, cnt);

  dim3 gC((NANCH + 255) / 256, BS);
  compactKernel<<<gC, 256, 0, stream>>>(score, iThresh, cnt, candIdx, candScoreC);

  sortKernel<<<BS, 1024, 0, stream>>>(cnt, candIdx, candScoreC, preds, clsArr,
                                      candScaled, candBoxOff, candScoreK, candClsK);

  nmsKernel<<<BS, 1024, 0, stream>>>(candBoxOff, candScoreK, candScaled, candClsK, out);
}